// G2_59889023975881
// MI455X (gfx1250) — compile-verified
//
#include <hip/hip_runtime.h>
#include <hip/hip_bf16.h>

typedef __attribute__((ext_vector_type(2))) float v2f;
typedef __attribute__((ext_vector_type(8))) float v8f;

#define DFEAT 64
#define ROWS_PER_BLOCK 128   // 8 wave32s x 16 rows
#define WT_PITCH 66          // transposed-W LDS pitch: even (b64 align), 2-way bank conflict on store only

// ---------------------------------------------------------------------------
// Zero a float range (graph-safe replacement for hipMemsetAsync bookkeeping).
// ---------------------------------------------------------------------------
__global__ void g2_zero_kernel(float* __restrict__ p, int n) {
    int i = blockIdx.x * blockDim.x + threadIdx.x;
    if (i < n) p[i] = 0.0f;
}

// ---------------------------------------------------------------------------
// Edge scatter #1: sums[dst] += nf[src], cnt[dst] += 1.
// 16 threads per edge, float4 per thread (64 floats/row). Gathers are
// coalesced per edge-row; atomics resolve in L2 (whole accumulator fits).
// ---------------------------------------------------------------------------
__global__ void g2_scatter_agg_kernel(const float* __restrict__ nf,
                                      const int* __restrict__ src,
                                      const int* __restrict__ dst,
                                      float* __restrict__ sums,
                                      float* __restrict__ cnt,
                                      int n_edges) {
    int gid = blockIdx.x * blockDim.x + threadIdx.x;
    int e = gid >> 4;
    if (e >= n_edges) return;
    int q = (gid & 15) * 4;                  // column quad 0..60
    int s = src[e];
    int d = dst[e];
    const float4 v = *(const float4*)(nf + (size_t)s * DFEAT + q);
    float* out = sums + (size_t)d * DFEAT + q;
    atomicAdd(out + 0, v.x);
    atomicAdd(out + 1, v.y);
    atomicAdd(out + 2, v.z);
    atomicAdd(out + 3, v.w);
    if (q == 0) atomicAdd(cnt + d, 1.0f);
}

// ---------------------------------------------------------------------------
// x = (sums / max(cnt,1)) @ W   via V_WMMA_F32_16X16X4_F32 (full fp32).
// Block: 256 threads = 8 wave32; each wave owns a 16-row slab and emits
// four 16x16 f32 tiles (N = 0,16,32,48). K=64 -> 16 chained WMMAs/tile.
// ISA f32 A/B fragment layout: lanes 0-15 hold K = 4k,4k+1; lanes 16-31
// hold K = 4k+2,4k+3 (M resp. N = lane & 15).
// C/D layout: VGPR v -> row v + 8*(lane>=16), col = lane & 15.
// W is staged TRANSPOSED in LDS so each lane's two B K-values are adjacent
// -> every A and B fragment is one aligned ds_load_b64 (no VGPR repacking).
// ---------------------------------------------------------------------------
__global__ __launch_bounds__(256) void g2_gemm_wmma_kernel(
        const float* __restrict__ sums,
        const float* __restrict__ cnt,
        const float* __restrict__ W,
        float* __restrict__ X,
        int n_nodes) {
    __shared__ float ldsA[ROWS_PER_BLOCK * DFEAT];   // 32 KB
    __shared__ float ldsWT[DFEAT * WT_PITCH];        // ~16.5 KB, WT[n*66 + k]

    const int tid = threadIdx.x;
    const int rowBase = blockIdx.x * ROWS_PER_BLOCK;

    // Stage W transposed: coalesced global read (k = i>>6, n = i&63),
    // LDS store bank pattern (2n+k)%64 -> 2-way conflict, one-time cost.
    for (int i = tid; i < DFEAT * DFEAT; i += 256) {
        int k = i >> 6, n = i & 63;
        ldsWT[n * WT_PITCH + k] = W[i];
    }

    // Stage agg slab, fusing the scatter_mean divide. OOB rows -> 0 so the
    // (EXEC-all-ones) WMMAs below chew on harmless data.
    for (int i = tid; i < ROWS_PER_BLOCK * DFEAT; i += 256) {
        int r = rowBase + (i >> 6);
        float v = 0.0f;
        if (r < n_nodes) {
            float c = cnt[r];
            v = sums[(size_t)r * DFEAT + (i & 63)] / fmaxf(c, 1.0f);
        }
        ldsA[i] = v;
    }
    __syncthreads();

    const int wave = tid >> 5;           // 0..7
    const int lane = tid & 31;
    const int m    = lane & 15;          // row (A) / col (B) within tile
    const int koff = (lane >> 4) << 1;   // 0 for lanes 0-15, 2 for 16-31

    const float* aRow = &ldsA[(wave * 16 + m) * DFEAT];
    const float* w0 = &ldsWT[(m +  0) * WT_PITCH];
    const float* w1 = &ldsWT[(m + 16) * WT_PITCH];
    const float* w2 = &ldsWT[(m + 32) * WT_PITCH];
    const float* w3 = &ldsWT[(m + 48) * WT_PITCH];

    v8f acc0 = {}, acc1 = {}, acc2 = {}, acc3 = {};
    for (int k = 0; k < 16; ++k) {
        const int kk = 4 * k + koff;     // even -> 8B-aligned b64 loads
        const v2f a  = *(const v2f*)(aRow + kk);
        const v2f b0 = *(const v2f*)(w0 + kk);
        const v2f b1 = *(const v2f*)(w1 + kk);
        const v2f b2 = *(const v2f*)(w2 + kk);
        const v2f b3 = *(const v2f*)(w3 + kk);
        acc0 = __builtin_amdgcn_wmma_f32_16x16x4_f32(false, a, false, b0,
                                                     (short)0, acc0, false, false);
        acc1 = __builtin_amdgcn_wmma_f32_16x16x4_f32(false, a, false, b1,
                                                     (short)0, acc1, false, false);
        acc2 = __builtin_amdgcn_wmma_f32_16x16x4_f32(false, a, false, b2,
                                                     (short)0, acc2, false, false);
        acc3 = __builtin_amdgcn_wmma_f32_16x16x4_f32(false, a, false, b3,
                                                     (short)0, acc3, false, false);
    }

    // Write back D tiles.
    const int row0 = rowBase + wave * 16 + ((lane >> 4) << 3);  // +0 or +8
    for (int v = 0; v < 8; ++v) {
        int row = row0 + v;
        if (row < n_nodes) {
            float* xr = X + (size_t)row * DFEAT + m;
            xr[0]  = acc0[v];
            xr[16] = acc1[v];
            xr[32] = acc2[v];
            xr[48] = acc3[v];
        }
    }
}

// ---------------------------------------------------------------------------
// Edge scatter #2: sums[dst] += (x[src]-x[dst])^2   (P == 2)
// ---------------------------------------------------------------------------
__global__ void g2_scatter_tau_kernel(const float* __restrict__ X,
                                      const int* __restrict__ src,
                                      const int* __restrict__ dst,
                                      float* __restrict__ sums,
                                      int n_edges) {
    int gid = blockIdx.x * blockDim.x + threadIdx.x;
    int e = gid >> 4;
    if (e >= n_edges) return;
    int q = (gid & 15) * 4;
    int s = src[e];
    int d = dst[e];
    const float4 a = *(const float4*)(X + (size_t)s * DFEAT + q);
    const float4 b = *(const float4*)(X + (size_t)d * DFEAT + q);
    float dx = a.x - b.x, dy = a.y - b.y, dz = a.z - b.z, dw = a.w - b.w;
    float* out = sums + (size_t)d * DFEAT + q;
    atomicAdd(out + 0, dx * dx);
    atomicAdd(out + 1, dy * dy);
    atomicAdd(out + 2, dz * dz);
    atomicAdd(out + 3, dw * dw);
}

// ---------------------------------------------------------------------------
// tau = tanh(sums / max(cnt, 1))
// ---------------------------------------------------------------------------
__global__ void g2_finalize_kernel(const float* __restrict__ sums,
                                   const float* __restrict__ cnt,
                                   float* __restrict__ out,
                                   int n_elems) {
    int i = blockIdx.x * blockDim.x + threadIdx.x;
    if (i >= n_elems) return;
    float c = fmaxf(cnt[i >> 6], 1.0f);
    out[i] = tanhf(sums[i] / c);
}

// ---------------------------------------------------------------------------
// Host-side launcher (graph-capture safe: kernels only, all on `stream`).
// Workspace layout (floats): [ sums 64N | cnt N | X 64N ]  -> 129N ~ 51.6 MB
// ---------------------------------------------------------------------------
extern "C" void kernel_launch(void* const* d_in, const int* in_sizes, int n_in,
                              void* d_out, int out_size, void* d_ws, size_t ws_size,
                              hipStream_t stream) {
    const float* nf = (const float*)d_in[0];       // [N, 64]
    const float* W  = (const float*)d_in[1];       // [64, 64]
    const int*   ei = (const int*)d_in[2];         // [2, E]
    float* out = (float*)d_out;                    // [N, 64]

    const int n_nodes = in_sizes[0] / DFEAT;
    const int n_edges = in_sizes[2] / 2;
    const int* src = ei;
    const int* dst = ei + n_edges;

    float* sums = (float*)d_ws;
    float* cnt  = sums + (size_t)n_nodes * DFEAT;
    float* X    = cnt + n_nodes;

    const int T = 256;

    // 1) zero sums + cnt (contiguous: 65N floats)
    {
        int n = n_nodes * (DFEAT + 1);
        g2_zero_kernel<<<(n + T - 1) / T, T, 0, stream>>>(sums, n);
    }
    // 2) scatter_mean accumulation of neighbor features
    {
        long long total = (long long)n_edges * 16;
        g2_scatter_agg_kernel<<<(int)((total + T - 1) / T), T, 0, stream>>>(
            nf, src, dst, sums, cnt, n_edges);
    }
    // 3) x = agg @ W  (fp32 WMMA)
    {
        int blocks = (n_nodes + ROWS_PER_BLOCK - 1) / ROWS_PER_BLOCK;
        g2_gemm_wmma_kernel<<<blocks, T, 0, stream>>>(sums, cnt, W, X, n_nodes);
    }
    // 4) re-zero sums for tau accumulation (cnt preserved — identical counts)
    {
        int n = n_nodes * DFEAT;
        g2_zero_kernel<<<(n + T - 1) / T, T, 0, stream>>>(sums, n);
    }
    // 5) scatter of squared feature differences
    {
        long long total = (long long)n_edges * 16;
        g2_scatter_tau_kernel<<<(int)((total + T - 1) / T), T, 0, stream>>>(
            X, src, dst, sums, n_edges);
    }
    // 6) tau = tanh(mean)
    {
        int n = n_nodes * DFEAT;
        g2_finalize_kernel<<<(n + T - 1) / T, T, 0, stream>>>(sums, cnt, out, n);
    }
}